// StdJacobiSGNN_7301444403243
// MI455X (gfx1250) — compile-verified
//
#include <hip/hip_runtime.h>
#include <cstdint>
#include <cmath>

#define TB 256
#define D_FEAT 48
#define SRC_MASK 0xFFFFF   // N must be < 2^20 (reference N = 100000)

typedef __attribute__((ext_vector_type(2))) float v2f;
typedef __attribute__((ext_vector_type(8))) float v8f;

struct NW11 { float v[11]; };

// ---------------- degree / norm ----------------
__global__ void k_deg(const long long* __restrict__ eidx, int* __restrict__ deg, int E) {
    int e = blockIdx.x * TB + threadIdx.x;
    if (e < E) {
        int d = (int)eidx[(size_t)E + (size_t)e];   // edge_index[1][e] = dst
        atomicAdd(&deg[d], 1);
    }
}

__global__ void k_dinv(const int* __restrict__ deg, float* __restrict__ dinv, int N) {
    int i = blockIdx.x * TB + threadIdx.x;
    if (i < N) {
        int dg = deg[i];
        dinv[i] = dg > 0 ? rsqrtf((float)dg) : 0.0f;
    }
}

// ---------------- exclusive scan of deg -> rowptr (3 phases) ----------------
__global__ void k_scan1(const int* __restrict__ deg, int* __restrict__ part, int N) {
    __shared__ int sh[TB];
    int i = blockIdx.x * TB + threadIdx.x;
    sh[threadIdx.x] = (i < N) ? deg[i] : 0;
    __syncthreads();
    for (int s = TB / 2; s > 0; s >>= 1) {
        if (threadIdx.x < s) sh[threadIdx.x] += sh[threadIdx.x + s];
        __syncthreads();
    }
    if (threadIdx.x == 0) part[blockIdx.x] = sh[0];
}

__global__ void k_scan2(int* __restrict__ part, int nb) {
    if (blockIdx.x == 0 && threadIdx.x == 0) {
        int run = 0;
        for (int i = 0; i < nb; i++) { int v = part[i]; part[i] = run; run += v; }
    }
}

__global__ void k_scan3(const int* __restrict__ deg, const int* __restrict__ part,
                        int* __restrict__ rowptr, int N, int E) {
    __shared__ int sh[TB];
    int t = threadIdx.x;
    int i = blockIdx.x * TB + t;
    int v = (i < N) ? deg[i] : 0;
    sh[t] = v;
    __syncthreads();
    for (int off = 1; off < TB; off <<= 1) {
        int u = (t >= off) ? sh[t - off] : 0;
        __syncthreads();
        sh[t] += u;
        __syncthreads();
    }
    if (i < N) rowptr[i] = sh[t] - v + part[blockIdx.x];
    if (i == 0) rowptr[N] = E;
}

// ---------------- CSR fill: pack src | (localdst << 20) ----------------
__global__ void k_fill(const long long* __restrict__ eidx, const float* __restrict__ dinv,
                       int* __restrict__ fill, const int* __restrict__ rowptr,
                       int* __restrict__ csr_sr, float* __restrict__ csr_w, int E) {
    int e = blockIdx.x * TB + threadIdx.x;
    if (e >= E) return;
    int s = (int)eidx[(size_t)e];
    int d = (int)eidx[(size_t)E + (size_t)e];
    int pos = rowptr[d] + atomicAdd(&fill[d], 1);
    csr_sr[pos] = s | ((d & 15) << 20);   // local row within 16-dst group
    csr_w[pos]  = dinv[s] * dinv[d];
}

// ---------------- per-k combined weights (device-side: tanh cumprod) ----------------
__global__ void k_coefs(const float* __restrict__ lap_coefs, const float* __restrict__ mf,
                        float* __restrict__ wc, NW11 nw) {
    if (blockIdx.x == 0 && threadIdx.x == 0) {
        const float alpha = 1.0f;
        wc[0] = mf[0] / nw.v[0];
        float cum = 1.0f;
        for (int i = 1; i <= 10; i++) {
            cum *= alpha * tanhf(lap_coefs[i - 1]);
            wc[i] = cum * mf[i] / nw.v[i];
        }
    }
}

// ---------------- fused WMMA propagation + Jacobi recurrence + accumulation --------
// One wave32 per 16 consecutive dst nodes. Edge chunks of 4 feed
// V_WMMA_F32_16X16X4_F32 with a one-hot scatter A (16x4) and gathered B (4x16),
// 3 feature tiles for D=48. Then per-lane:
//   v = alpha*prop + beta*Pm1 + gamma*Pm2 ; Pn = v ;
//   acc = init ? wc[0]*Pm1 + wc[n]*v : acc + wc[n]*v
__global__ void k_prop_wmma(const float* __restrict__ h,
                            const int* __restrict__ rowptr,
                            const int* __restrict__ csr_sr,
                            const float* __restrict__ csr_w,
                            const float* __restrict__ pm1,
                            const float* __restrict__ pm2,
                            float* __restrict__ pn,
                            float* __restrict__ acc,
                            const float* __restrict__ wc,
                            int n, int init,
                            float alpha, float beta, float gamma, int N) {
    const int wave  = blockIdx.x * (TB / 32) + (threadIdx.x >> 5);
    const int gbase = wave * 16;
    if (gbase >= N) return;                 // wave-uniform: EXEC all-ones below
    const int l  = threadIdx.x & 31;
    const int hi = l >> 4;
    const int ml = l & 15;

    int gend = gbase + 16; if (gend > N) gend = N;
    // Bounds are wave-uniform: force into SGPRs for scalar loop control and
    // to guarantee EXEC == all-ones through the WMMA region.
    const int beg = __builtin_amdgcn_readfirstlane(rowptr[gbase]);
    const int end = __builtin_amdgcn_readfirstlane(rowptr[gend]);

    v8f c0 = {}, c1 = {}, c2 = {};

    for (int kb = beg; kb < end; kb += 4) {
        // Speculative near-cache prefetch of the edge stream (non-faulting;
        // overshoot stays inside the workspace).
        __builtin_prefetch((const void*)(csr_sr + kb + 32), 0, 3);
        __builtin_prefetch((const void*)(csr_w  + kb + 32), 0, 3);

        // A-matrix K layout (ISA 7.12.2): vgpr0 <-> K = hi?2:0 ; vgpr1 <-> K = hi?3:1
        int k0 = kb + (hi ? 2 : 0);
        int k1 = kb + (hi ? 3 : 1);
        int kc0 = k0 < end ? k0 : end - 1;  // clamp: pad lanes read a valid (finite) row
        int kc1 = k1 < end ? k1 : end - 1;
        int   p0  = csr_sr[kc0];            // unconditional loads (in-bounds by clamp),
        int   p1  = csr_sr[kc1];            // zero-select afterwards -> no exec branches
        float w0r = csr_w[kc0];
        float w1r = csr_w[kc1];
        float w0 = (k0 < end) ? w0r : 0.0f; // pad weight = 0
        float w1 = (k1 < end) ? w1r : 0.0f;
        int s0 = p0 & SRC_MASK;
        int s1 = p1 & SRC_MASK;
        int d0 = (int)(((unsigned)p0) >> 20);
        int d1 = (int)(((unsigned)p1) >> 20);

        v2f a;
        a.x = (d0 == ml) ? w0 : 0.0f;       // one-hot scatter column
        a.y = (d1 == ml) ? w1 : 0.0f;

        const float* r0 = h + (size_t)s0 * D_FEAT + ml;
        const float* r1 = h + (size_t)s1 * D_FEAT + ml;

        v2f b;
        b.x = r0[0];   b.y = r1[0];
        c0 = __builtin_amdgcn_wmma_f32_16x16x4_f32(false, a, false, b, (short)0, c0, false, false);
        b.x = r0[16];  b.y = r1[16];
        c1 = __builtin_amdgcn_wmma_f32_16x16x4_f32(false, a, false, b, (short)0, c1, false, false);
        b.x = r0[32];  b.y = r1[32];
        c2 = __builtin_amdgcn_wmma_f32_16x16x4_f32(false, a, false, b, (short)0, c2, false, false);
    }

    const float wn  = wc[n];
    const float w0i = wc[0];
    #pragma unroll
    for (int j = 0; j < 8; ++j) {
        int row = gbase + j + hi * 8;       // C layout: vgpr j <-> M = j + 8*hi, N = ml
        if (row < N) {
            size_t base = (size_t)row * D_FEAT + ml;
            #pragma unroll
            for (int t = 0; t < 3; ++t) {
                float cv = (t == 0) ? c0[j] : (t == 1) ? c1[j] : c2[j];
                size_t idx = base + (size_t)t * 16;
                float m1 = pm1[idx];
                float v  = alpha * cv + beta * m1 + gamma * pm2[idx];
                pn[idx]  = v;
                acc[idx] = init ? (w0i * m1 + wn * v) : (acc[idx] + wn * v);
            }
        }
    }
}

extern "C" void kernel_launch(void* const* d_in, const int* in_sizes, int n_in,
                              void* d_out, int out_size, void* d_ws, size_t ws_size,
                              hipStream_t stream) {
    (void)n_in; (void)out_size; (void)ws_size;
    const float*     x         = (const float*)d_in[0];
    const long long* eidx      = (const long long*)d_in[1];
    const float*     lap_coefs = (const float*)d_in[2];
    const float*     mf        = (const float*)d_in[3];

    const int N  = in_sizes[0] / D_FEAT;
    const int E  = in_sizes[1] / 2;
    const size_t M = (size_t)N * D_FEAT;

    // ---- workspace carve-out (256B aligned segments) ----
    char* p = (char*)d_ws;
    auto alloc = [&](size_t bytes) -> char* {
        char* r = p;
        p += (bytes + 255) & ~(size_t)255;
        return r;
    };
    const int nb = (N + TB - 1) / TB;
    int*   deg    = (int*)  alloc((size_t)N * 4);
    int*   rowptr = (int*)  alloc(((size_t)N + 1) * 4);
    int*   fill   = (int*)  alloc((size_t)N * 4);
    int*   part   = (int*)  alloc((size_t)nb * 4);
    float* dinv   = (float*)alloc((size_t)N * 4);
    int*   csr_sr = (int*)  alloc((size_t)E * 4);
    float* csr_w  = (float*)alloc((size_t)E * 4);
    float* wc     = (float*)alloc(256);
    float* buf0   = (float*)alloc(M * 4);
    float* buf1   = (float*)alloc(M * 4);
    float* buf2   = (float*)alloc(M * 4);

    (void)hipMemsetAsync(deg,  0, (size_t)N * 4, stream);
    (void)hipMemsetAsync(fill, 0, (size_t)N * 4, stream);

    // ---- host-side Jacobi constants (a = b = 1) ----
    const double a = 1.0, b = 1.0;
    NW11 nwf;
    for (int i = 0; i <= 10; i++) {
        double t1 = pow(2.0, a + b + 1.0) / (2.0 * i + a + b + 1.0);
        double t2 = exp(lgamma(i + a + 1.0) - lgamma(i + a + b + 1.0));
        double t3 = exp(lgamma(i + b + 1.0) - lgamma(i + 1.0));
        nwf.v[i] = (float)sqrt(t1 * t2 * t3);
    }
    const double c0 = (a - b) / 2.0;
    const double c1 = (a + b + 2.0) / 2.0;

    const int gE = (E + TB - 1) / TB;
    const int gN = (N + TB - 1) / TB;
    const int ngroups = (N + 15) / 16;                 // one wave32 per group
    const int gW = (ngroups + (TB / 32) - 1) / (TB / 32);

    // ---- build normalized CSR-by-destination (amortized over 10 props) ----
    k_deg  <<<gE, TB, 0, stream>>>(eidx, deg, E);
    k_dinv <<<gN, TB, 0, stream>>>(deg, dinv, N);
    k_scan1<<<gN, TB, 0, stream>>>(deg, part, N);
    k_scan2<<<1, 32, 0, stream>>>(part, nb);
    k_scan3<<<gN, TB, 0, stream>>>(deg, part, rowptr, N, E);
    k_fill <<<gE, TB, 0, stream>>>(eidx, dinv, fill, rowptr, csr_sr, csr_w, E);
    k_coefs<<<1, 32, 0, stream>>>(lap_coefs, mf, wc, nwf);

    float* accp = (float*)d_out;

    // ---- P1 = c0*x + c1*prop(x); acc = wc0*x + wc1*P1 ----
    k_prop_wmma<<<gW, TB, 0, stream>>>(x, rowptr, csr_sr, csr_w,
                                       x, x, buf0, accp, wc,
                                       1, 1, (float)c1, (float)c0, 0.0f, N);

    // ---- P2..P10 with rotating buffers ----
    const float* Pm2 = x;
    float*       Pm1 = buf0;
    float* bufs[3] = { buf0, buf1, buf2 };
    int bi = 1;
    for (int n = 2; n <= 10; n++) {
        double dn  = (double)n;
        double nab = 2.0 * dn + a + b;
        double den = 2.0 * dn * (nab - dn) * (nab - 2.0);
        double an  = nab * (nab - 1.0) * (nab - 2.0) / den;
        double bn  = (nab - 1.0) * (a * a - b * b) / den;
        double cn  = -(2.0 * (dn + a - 1.0) * (dn + b - 1.0) * nab) / den;

        float* Pn = bufs[bi];
        bi = (bi + 1) % 3;
        k_prop_wmma<<<gW, TB, 0, stream>>>(Pm1, rowptr, csr_sr, csr_w,
                                           Pm1, Pm2, Pn, accp, wc,
                                           n, 0, (float)an, (float)bn, (float)cn, N);
        Pm2 = Pm1;
        Pm1 = Pn;
    }
}